// LSM_45999099740486
// MI455X (gfx1250) — compile-verified
//
#include <hip/hip_runtime.h>
#include <math.h>

// ---------------------------------------------------------------------------
// MI455X (gfx1250) latent-space-model likelihood.
//
// Roofline: 6M pairs * 128B gathered row data ~= 0.8GB logical traffic from a
// 9.6MB working set -> fully L2-resident, gather/latency bound; ~0.4 GFLOP of
// VALU is noise. So: one pair per lane, 4x global_load_b128 per 16-float row,
// per-lane FMA reduction, sqrt (+exp). WMMA is used where it is free: the
// wave-level 32-lane sum via V_WMMA_F32_16X16X4_F32 (A = lane values in
// VGPR0/K={0,2}, VGPR1 = 0; B = all-ones so D[m,n] = x[m]+x[m+16]).
// ---------------------------------------------------------------------------

typedef __attribute__((ext_vector_type(2))) float v2f;
typedef __attribute__((ext_vector_type(8))) float v8f;

// Wave32 sum via the CDNA5 matrix pipe. Requires EXEC == all ones.
__device__ __forceinline__ float wave_sum_wmma(float x) {
    v2f a; a[0] = x;    a[1] = 0.0f;   // A[m,0]=x(lane m), A[m,2]=x(lane m+16), K=1,3 -> 0
    v2f b; b[0] = 1.0f; b[1] = 1.0f;   // all-ones B: layout-agnostic
    v8f c = {};
    // D[m,n] = x[m] + x[m+16] for every n   (v_wmma_f32_16x16x4_f32)
    v8f d = __builtin_amdgcn_wmma_f32_16x16x4_f32(
        false, a, false, b, (short)0, c, false, false);
    // lanes 0-15 hold rows M=0..7, lanes 16-31 hold rows M=8..15
    float s = d[0] + d[1] + d[2] + d[3] + d[4] + d[5] + d[6] + d[7];
    s += __shfl_xor(s, 16, 32);        // combine the two half-sums
    return s;                          // full 32-lane sum in every lane
}

// Block reduction: WMMA wave sums -> LDS -> one f32 atomic per block.
// Every thread of the block must call this (no early exits in callers).
__device__ __forceinline__ void block_accum(float v, float* __restrict__ gout) {
    __shared__ float lds[8];           // 256 threads = 8 waves (wave32)
    float ws = wave_sum_wmma(v);
    const int wave = threadIdx.x >> 5;
    const int lane = threadIdx.x & 31;
    if (lane == 0) lds[wave] = ws;
    __syncthreads();
    if (threadIdx.x == 0) {
        float t = 0.0f;
        const int nw = blockDim.x >> 5;
        #pragma unroll
        for (int i = 0; i < 8; ++i) if (i < nw) t += lds[i];
        atomicAdd(gout, t);            // global_atomic_add_f32
    }
}

__global__ void lsm_zero_acc(float* __restrict__ acc) {
    if (threadIdx.x < 2) acc[threadIdx.x] = 0.0f;
}

// sum over edges of  -sqrt( sum_d (z[i,d]-w[j,d]+1e-6)^2 )
__global__ void lsm_sparse_term(const float* __restrict__ z,
                                const float* __restrict__ w,
                                const int*   __restrict__ ii,
                                const int*   __restrict__ jj,
                                int n, float* __restrict__ acc) {
    const int e = blockIdx.x * blockDim.x + threadIdx.x;
    float local = 0.0f;
    if (e < n) {
        const float4* zr = (const float4*)z + (size_t)ii[e] * 4;  // 4x b128
        const float4* wr = (const float4*)w + (size_t)jj[e] * 4;
        float s = 0.0f;
        #pragma unroll
        for (int c = 0; c < 4; ++c) {
            float4 a = zr[c];
            float4 b = wr[c];
            float d0 = a.x - b.x + 1e-6f;
            float d1 = a.y - b.y + 1e-6f;
            float d2 = a.z - b.z + 1e-6f;
            float d3 = a.w - b.w + 1e-6f;
            s = fmaf(d0, d0, s); s = fmaf(d1, d1, s);
            s = fmaf(d2, d2, s); s = fmaf(d3, d3, s);
        }
        local = -sqrtf(s);
    }
    block_accum(local, acc);           // all lanes active: EXEC==~0 at WMMA
}

// sum over pairs of  exp( b - (sqrt(sum_d (z[i,d]-w[j,d])^2) + 1e-8) )
__global__ void lsm_analytic_term(const float* __restrict__ z,
                                  const float* __restrict__ w,
                                  const int*   __restrict__ ii,
                                  const int*   __restrict__ jj,
                                  int n,
                                  const float* __restrict__ bias,
                                  float* __restrict__ acc) {
    const float b0 = bias[0];          // uniform scalar load
    const int e = blockIdx.x * blockDim.x + threadIdx.x;
    float local = 0.0f;
    if (e < n) {
        const float4* zr = (const float4*)z + (size_t)ii[e] * 4;
        const float4* wr = (const float4*)w + (size_t)jj[e] * 4;
        float s = 0.0f;
        #pragma unroll
        for (int c = 0; c < 4; ++c) {
            float4 a = zr[c];
            float4 b = wr[c];
            float d0 = a.x - b.x;
            float d1 = a.y - b.y;
            float d2 = a.z - b.z;
            float d3 = a.w - b.w;
            s = fmaf(d0, d0, s); s = fmaf(d1, d1, s);
            s = fmaf(d2, d2, s); s = fmaf(d3, d3, s);
        }
        const float dist = sqrtf(s) + 1e-8f;
        local = expf(b0 - dist);
    }
    block_accum(local, acc);
}

// out = sparse_acc + E*b - theta - analytic_acc
__global__ void lsm_finalize(const float* __restrict__ acc,
                             const float* __restrict__ bias,
                             const float* __restrict__ thetas,
                             float Ef, float* __restrict__ out) {
    out[0] = acc[0] + Ef * bias[0] - thetas[0] - acc[1];
}

extern "C" void kernel_launch(void* const* d_in, const int* in_sizes, int n_in,
                              void* d_out, int out_size, void* d_ws, size_t ws_size,
                              hipStream_t stream) {
    const float* z      = (const float*)d_in[0];  // latent_z  [100000,16]
    const float* w      = (const float*)d_in[1];  // latent_w  [50000,16]
    const float* bias   = (const float*)d_in[2];  // [1]
    const float* thetas = (const float*)d_in[3];  // [1]
    const int*   si     = (const int*)d_in[4];    // sparse_i  [E]
    const int*   sj     = (const int*)d_in[5];    // sparse_j  [E]
    const int*   ai     = (const int*)d_in[6];    // analytical_i [M]
    const int*   aj     = (const int*)d_in[7];    // analytical_j [M]

    const int E = in_sizes[4];
    const int M = in_sizes[6];

    float* acc = (float*)d_ws;                    // acc[0]=sparse, acc[1]=analytic

    lsm_zero_acc<<<1, 32, 0, stream>>>(acc);

    const int BS = 256;                           // 8 wave32 waves per block
    lsm_sparse_term  <<<(E + BS - 1) / BS, BS, 0, stream>>>(z, w, si, sj, E, acc + 0);
    lsm_analytic_term<<<(M + BS - 1) / BS, BS, 0, stream>>>(z, w, ai, aj, M, bias, acc + 1);

    lsm_finalize<<<1, 1, 0, stream>>>(acc, bias, thetas, (float)E, (float*)d_out);
}